// MyModel_61933428409488
// MI455X (gfx1250) — compile-verified
//
#include <hip/hip_runtime.h>
#include <hip/hip_bf16.h>
#include <stdint.h>

// ---------------------------------------------------------------------------
// LSTM, B=8192, T=2048, D=2, H=2.  One lane per batch row; x staged through
// LDS via the CDNA5 Tensor Data Mover (tensor_load_to_lds), double buffered
// on TENSORcnt.  All activations via the CDNA5 hardware v_tanh_f32:
//   sigmoid(a) = 0.5 + 0.5*tanh(a/2), with the /2 folded into the (uniform)
//   gate weights for i/f/o so it costs nothing per step.
// ---------------------------------------------------------------------------

typedef unsigned int u32x4 __attribute__((ext_vector_type(4)));
typedef int          i32x4 __attribute__((ext_vector_type(4)));
typedef int          i32x8 __attribute__((ext_vector_type(8)));

static constexpr int   T_LEN  = 2048;
static constexpr int   NB     = 32;                // batch rows per workgroup (= wave32)
static constexpr int   NT     = 64;                // timesteps per LDS tile
static constexpr int   NCH    = T_LEN / NT;        // 32 chunks
static constexpr int   ROW_DW = NT * 2 + 2;        // 130 dwords: 512B row + 8B pad

__device__ __forceinline__ float htanh(float x) {
    return __builtin_amdgcn_tanhf(x);              // v_tanh_f32
}

// Issue one TDM 2D tile load: NB rows x (NT*8B) from global (row stride
// T_LEN*8B) into LDS at lds_byte, padding +8B after every 512B row so the
// LDS row stride is 130 dwords (conflict-free ds_load_b64 reads).
__device__ __forceinline__ void tdm_load_tile(const float* gptr, unsigned lds_byte) {
    unsigned long long ga = (unsigned long long)(uintptr_t)gptr;
    u32x4 g0;
    g0[0] = 1u;                                            // count=1 (valid user D#)
    g0[1] = lds_byte;                                      // LDS byte address
    g0[2] = (unsigned)ga;                                  // global_addr[31:0]
    g0[3] = ((unsigned)(ga >> 32) & 0x01ffffffu)           // global_addr[56:32]
          | (2u << 30);                                    // type = 2 ("image")
    i32x8 g1;
    g1[0] = (3 << 16)                                      // data_size = 8 bytes
          | (1 << 20)                                      // pad_enable
          | (6 << 22)                                      // pad_interval: 512B
          | (1 << 25);                                     // pad_amount: 2 dwords
    g1[1] = (NT & 0xffff) << 16;                           // tensor_dim0 = NT (8B units)
    g1[2] = (NB & 0xffff) << 16;                           // tensor_dim1 = NB
    g1[3] = (NT & 0xffff) << 16;                           // tile_dim0 = NT
    g1[4] = (NB & 0xffff);                                 // tile_dim1 = NB, tile_dim2 = 0
    g1[5] = T_LEN;                                         // tensor_dim0_stride (8B units)
    g1[6] = 0;                                             // stride0 hi / stride1 lo
    g1[7] = 0;                                             // stride1 hi
    i32x4 z4 = {0, 0, 0, 0};
    i32x8 z8 = {0, 0, 0, 0, 0, 0, 0, 0};
    __builtin_amdgcn_tensor_load_to_lds(g0, g1, z4, z4, z8, 0);
}

__global__ __launch_bounds__(NB) void lstm_tdm_kernel(
    const float* __restrict__ x,     // (B, T, 2)
    const float* __restrict__ h0,    // (B, 2)
    const float* __restrict__ c0,    // (B, 2)
    const float* __restrict__ w_ih,  // (8, 2)
    const float* __restrict__ w_hh,  // (8, 2)
    const float* __restrict__ b_ih,  // (8)
    const float* __restrict__ b_hh,  // (8)
    float* __restrict__ out)         // (B, 2) final cell state
{
    __shared__ __align__(16) float tile[2][NB * ROW_DW];

    const int lane = threadIdx.x;
    const int b    = blockIdx.x * NB + lane;

    // Uniform weights -> SGPRs; pre-sum biases.  Gates 0,1 (i), 2,3 (f) and
    // 6,7 (o) feed sigmoid(a) = 0.5 + 0.5*tanh(a/2): pre-scale their rows by
    // 0.5.  Gates 4,5 (g) feed tanh(a) directly: unscaled.
    float wi0[8], wi1[8], wh0[8], wh1[8], bs[8];
#pragma unroll
    for (int g = 0; g < 8; ++g) {
        const float s = (g == 4 || g == 5) ? 1.0f : 0.5f;
        wi0[g] = s * w_ih[2 * g];
        wi1[g] = s * w_ih[2 * g + 1];
        wh0[g] = s * w_hh[2 * g];
        wh1[g] = s * w_hh[2 * g + 1];
        bs[g]  = s * (b_ih[g] + b_hh[g]);
    }

    float hv0 = h0[2 * b], hv1 = h0[2 * b + 1];
    float cv0 = c0[2 * b], cv1 = c0[2 * b + 1];

    const float* xblk = x + (size_t)blockIdx.x * NB * (T_LEN * 2);
    unsigned lds0 = (unsigned)(uintptr_t)&tile[0][0];
    unsigned lds1 = (unsigned)(uintptr_t)&tile[1][0];

    tdm_load_tile(xblk, lds0);  // prefetch chunk 0

    for (int ch = 0; ch < NCH; ++ch) {
        if (ch + 1 < NCH) {
            tdm_load_tile(xblk + (size_t)(ch + 1) * NT * 2,
                          ((ch + 1) & 1) ? lds1 : lds0);
            __builtin_amdgcn_s_wait_tensorcnt(1);  // in-order: current tile done
        } else {
            __builtin_amdgcn_s_wait_tensorcnt(0);
        }

        const float2* row = (const float2*)&tile[ch & 1][lane * ROW_DW];
#pragma unroll 4
        for (int t = 0; t < NT; ++t) {
            float2 xv = row[t];  // ds_load_b64, bank-conflict free
            float a[8];
#pragma unroll
            for (int g = 0; g < 8; ++g)
                a[g] = fmaf(wi0[g], xv.x,
                        fmaf(wi1[g], xv.y,
                         fmaf(wh0[g], hv0,
                          fmaf(wh1[g], hv1, bs[g]))));
            // torch gate order: i, f, g, o  (i/f/o pre-scaled by 1/2)
            float ig0 = fmaf(0.5f, htanh(a[0]), 0.5f);
            float ig1 = fmaf(0.5f, htanh(a[1]), 0.5f);
            float fg0 = fmaf(0.5f, htanh(a[2]), 0.5f);
            float fg1 = fmaf(0.5f, htanh(a[3]), 0.5f);
            float gg0 = htanh(a[4]);
            float gg1 = htanh(a[5]);
            float og0 = fmaf(0.5f, htanh(a[6]), 0.5f);
            float og1 = fmaf(0.5f, htanh(a[7]), 0.5f);
            cv0 = fmaf(fg0, cv0, ig0 * gg0);
            cv1 = fmaf(fg1, cv1, ig1 * gg1);
            hv0 = og0 * htanh(cv0);
            hv1 = og1 * htanh(cv1);
        }
    }

    out[2 * b]     = cv0;
    out[2 * b + 1] = cv1;
}

extern "C" void kernel_launch(void* const* d_in, const int* in_sizes, int n_in,
                              void* d_out, int out_size, void* d_ws, size_t ws_size,
                              hipStream_t stream) {
    const float* x    = (const float*)d_in[0];
    const float* h0   = (const float*)d_in[1];
    const float* c0   = (const float*)d_in[2];
    const float* w_ih = (const float*)d_in[3];
    const float* w_hh = (const float*)d_in[4];
    const float* b_ih = (const float*)d_in[5];
    const float* b_hh = (const float*)d_in[6];
    float* out = (float*)d_out;

    const int B = in_sizes[1] / 2;  // h0 has B*H elements, H=2
    dim3 grid(B / NB);              // 8192/32 = 256 single-wave workgroups
    dim3 block(NB);
    lstm_tdm_kernel<<<grid, block, 0, stream>>>(x, h0, c0, w_ih, w_hh, b_ih, b_hh, out);
}